// Vit_Spatiochannel_Map_1992864825606
// MI455X (gfx1250) — compile-verified
//
#include <hip/hip_runtime.h>
#include <math.h>

// Problem constants (from reference)
#define BN     8        // B*N
#define CCH    64       // channels
#define IMG    128
#define PS     8
#define HP     16       // IMG/PS
#define NPATCH 256      // HP*HP
#define NPC    16384    // NPATCH*CCH tokens per batch item
#define DIM    64
#define HEADS  2
#define DH     32
#define QROWS  16       // BN*HEADS
#define GP     68       // gate-kernel LDS pitch: 68%64==4 -> conflict-free, 16B-aligned rows

typedef __attribute__((ext_vector_type(2))) float v2f;
typedef __attribute__((ext_vector_type(8))) float v8f;

__device__ __forceinline__ v8f wmma4(v2f a, v2f b, v8f c) {
  // D = A(16x4) * B(4x16) + C(16x16), fp32
  return __builtin_amdgcn_wmma_f32_16x16x4_f32(false, a, false, b, (short)0, c, false, false);
}

// CDNA5 async DMA: 16-byte copy from global vaddr to LDS offset vdst, tracked by ASYNCcnt
__device__ __forceinline__ void async_b128(unsigned ldsoff, const float* g) {
  asm volatile("global_load_async_to_lds_b128 %0, %1, off"
               :: "v"(ldsoff), "v"(g) : "memory");
}
__device__ __forceinline__ void wait_async0() {
  asm volatile("s_wait_asynccnt 0x0" ::: "memory");
}

// ---------------------------------------------------------------------------
// K0: Wq_avg[h][d] = mean over head rows of W_q  (2 x 64)
__global__ void k_wqavg(const float* __restrict__ Wq, float* __restrict__ WqA) {
  int tid = threadIdx.x;
  if (tid < HEADS * DIM) {
    int h = tid >> 6, d = tid & 63;
    float s = 0.f;
    for (int r = 0; r < DH; ++r) s += Wq[(h * DH + r) * DIM + d];
    WqA[tid] = s * (1.f / DH);
  }
}

// ---------------------------------------------------------------------------
// K1: xe = patches @ W_embed^T + b_embed + pos   (WMMA f32 16x16x4)
__global__ __launch_bounds__(128)
void k_embed(const float* __restrict__ x, const float* __restrict__ Wemb,
             const float* __restrict__ bemb, const float* __restrict__ pos,
             float* __restrict__ xe) {
  __shared__ float lds_a[64 * 65];   // patches[c][k], pitch 65
  __shared__ float lds_b[64 * 65];   // W_embed^T : [k][n]
  const int tid  = threadIdx.x;
  const int bn   = blockIdx.x >> 8;
  const int pidx = blockIdx.x & 255;
  const int py = pidx >> 4, px = pidx & 15;

  const float* xb = x + (size_t)bn * CCH * IMG * IMG;
  for (int s = 0; s < 32; ++s) {                 // 4096 patch elems
    int idx = s * 128 + tid;
    int c = idx >> 6, k = idx & 63;
    int row = py * PS + (k >> 3), col = px * PS + (k & 7);
    lds_a[c * 65 + k] = xb[((size_t)c * IMG + row) * IMG + col];
  }
  for (int s = 0; s < 32; ++s) {                 // W_embed transposed
    int idx = s * 128 + tid;
    int n = idx >> 6, k = idx & 63;
    lds_b[k * 65 + n] = Wemb[n * 64 + k];
  }
  __syncthreads();

  const int w = tid >> 5, lane = tid & 31;
  const int half = lane >> 4, l16 = lane & 15;
  const int arow = (w * 16 + l16) * 65;
  v8f acc[4] = {{}, {}, {}, {}};
  for (int kc = 0; kc < 16; ++kc) {
    int kk = kc * 4 + half * 2;
    v2f a; a.x = lds_a[arow + kk]; a.y = lds_a[arow + kk + 1];
#pragma unroll
    for (int nt = 0; nt < 4; ++nt) {
      v2f b; b.x = lds_b[kk * 65 + nt * 16 + l16];
      b.y = lds_b[(kk + 1) * 65 + nt * 16 + l16];
      acc[nt] = wmma4(a, b, acc[nt]);
    }
  }
  const int tb = pidx * 64 + w * 16;             // token base within bn
#pragma unroll
  for (int nt = 0; nt < 4; ++nt) {
#pragma unroll
    for (int r = 0; r < 8; ++r) {
      int m = r + half * 8;
      int n = nt * 16 + l16;
      int t = tb + m;
      xe[((size_t)bn * NPC + t) * DIM + n] =
          acc[nt][r] + bemb[n] + pos[(size_t)t * DIM + n];
    }
  }
}

// ---------------------------------------------------------------------------
// K2: q[(bn*2+h)][t] = dot(xe[bn,t,:], Wq_avg[h,:])
__global__ __launch_bounds__(256)
void k_q(const float* __restrict__ xe, const float* __restrict__ WqA,
         float* __restrict__ Q) {
  __shared__ float wq[128];
  int tid = threadIdx.x;
  if (tid < 128) wq[tid] = WqA[tid];
  __syncthreads();
  int gid = blockIdx.x * 256 + tid;              // 0 .. BN*NPC-1
  int bn = gid >> 14;
  int t  = gid & (NPC - 1);
  const float* row = xe + (size_t)gid * DIM;
  float a0 = 0.f, a1 = 0.f;
  for (int d = 0; d < DIM; ++d) {
    float xv = row[d];
    a0 += xv * wq[d];
    a1 += xv * wq[64 + d];
  }
  Q[(size_t)(bn * 2 + 0) * NPC + t] = a0;
  Q[(size_t)(bn * 2 + 1) * NPC + t] = a1;
}

// ---------------------------------------------------------------------------
// K3: KG = sigmoid(Q(16x16384) @ W_k^T) — the 1.07 GB bandwidth-bound GEMM.
// Double-buffered GLOBAL_LOAD_ASYNC_TO_LDS_B128 pipeline: async DMA of the
// next 64x64 W_k chunk (+16x64 Q chunk) overlaps the 16 WMMAs on the current
// chunk; one s_wait_asynccnt + one barrier per chunk.
__global__ __launch_bounds__(128)
void k_gate(const float* __restrict__ Q, const float* __restrict__ Wk,
            float* __restrict__ KG) {
  __shared__ float lds_w[2][64 * GP];
  __shared__ float lds_q[2][16 * GP];
  const int tid = threadIdx.x;
  const int w = tid >> 5, lane = tid & 31;
  const int half = lane >> 4, l16 = lane & 15;
  const int jbase = blockIdx.x * 64;
  const int j0 = jbase + w * 16;

  // LDS byte offsets of the two buffers (flat addr low 32 bits == LDS offset)
  unsigned lw[2] = {(unsigned)(size_t)&lds_w[0][0], (unsigned)(size_t)&lds_w[1][0]};
  unsigned lq[2] = {(unsigned)(size_t)&lds_q[0][0], (unsigned)(size_t)&lds_q[1][0]};

  // per-thread float4 slots: W_k tile 1024 float4 (8/thread), Q tile 256 (2/thread)
  auto stage = [&](int kb, unsigned lwb, unsigned lqb) {
#pragma unroll
    for (int s = 0; s < 8; ++s) {
      int u = s * 128 + tid;
      int r = u >> 4, c4 = u & 15;
      async_b128(lwb + (unsigned)(r * GP + c4 * 4) * 4u,
                 Wk + (size_t)(jbase + r) * NPC + kb * 64 + c4 * 4);
    }
#pragma unroll
    for (int s = 0; s < 2; ++s) {
      int u = s * 128 + tid;
      int r = u >> 4, c4 = u & 15;
      async_b128(lqb + (unsigned)(r * GP + c4 * 4) * 4u,
                 Q + (size_t)r * NPC + kb * 64 + c4 * 4);
    }
  };

  v8f acc = {};
  stage(0, lw[0], lq[0]);
  wait_async0();
  __syncthreads();

  const int NCH = NPC / 64;
  for (int kb = 0; kb < NCH; ++kb) {
    int p = kb & 1;
    if (kb + 1 < NCH) stage(kb + 1, lw[p ^ 1], lq[p ^ 1]);
    const float* bw = &lds_w[p][0];
    const float* bq = &lds_q[p][0];
#pragma unroll
    for (int kc = 0; kc < 16; ++kc) {
      int kk = kc * 4 + half * 2;
      v2f a; a.x = bq[l16 * GP + kk];            a.y = bq[l16 * GP + kk + 1];
      v2f b; b.x = bw[(w * 16 + l16) * GP + kk]; b.y = bw[(w * 16 + l16) * GP + kk + 1];
      acc = wmma4(a, b, acc);
    }
    wait_async0();
    __syncthreads();
  }

#pragma unroll
  for (int r = 0; r < 8; ++r) {
    int m = r + half * 8;                        // q-row (bn*2+h)
    float vv = acc[r];
    vv = 1.f / (1.f + __expf(-vv));
    KG[(size_t)m * NPC + j0 + l16] = vv;
  }
}

// ---------------------------------------------------------------------------
// K4: v = xe @ W_v^T; g = gate * v; out = g @ W_out^T + b_out; fold-scatter.
__global__ __launch_bounds__(128)
void k_out(const float* __restrict__ xe, const float* __restrict__ Wv,
           const float* __restrict__ Wo, const float* __restrict__ bo,
           const float* __restrict__ KG, float* __restrict__ out) {
  __shared__ float lds_xe[64 * 65];
  __shared__ float lds_wv[64 * 65];  // W_v^T  : [e][d]
  __shared__ float lds_wo[64 * 65];  // W_out^T: [d][p]
  __shared__ float lds_g[4 * 16 * 65];
  __shared__ float lds_kg[128];
  const int tid  = threadIdx.x;
  const int bn   = blockIdx.x >> 8;
  const int pidx = blockIdx.x & 255;
  const int py = pidx >> 4, px = pidx & 15;

  for (int s = 0; s < 32; ++s) {
    int idx = s * 128 + tid;
    int c = idx >> 6, e = idx & 63;
    lds_xe[c * 65 + e] = xe[((size_t)bn * NPC + pidx * 64 + c) * DIM + e];
  }
  for (int s = 0; s < 32; ++s) {
    int idx = s * 128 + tid;
    int d = idx >> 6, e = idx & 63;
    lds_wv[e * 65 + d] = Wv[d * 64 + e];
    lds_wo[e * 65 + d] = Wo[d * 64 + e];
  }
  if (tid < 128) {
    int h = tid >> 6, c = tid & 63;
    lds_kg[tid] = KG[(size_t)(bn * 2 + h) * NPC + pidx * 64 + c];
  }
  __syncthreads();

  const int w = tid >> 5, lane = tid & 31;
  const int half = lane >> 4, l16 = lane & 15;
  const int arow = (w * 16 + l16) * 65;

  // v-tile (16 tokens x 64 dims)
  v8f accv[4] = {{}, {}, {}, {}};
  for (int kc = 0; kc < 16; ++kc) {
    int kk = kc * 4 + half * 2;
    v2f a; a.x = lds_xe[arow + kk]; a.y = lds_xe[arow + kk + 1];
#pragma unroll
    for (int nt = 0; nt < 4; ++nt) {
      v2f b; b.x = lds_wv[kk * 65 + nt * 16 + l16];
      b.y = lds_wv[(kk + 1) * 65 + nt * 16 + l16];
      accv[nt] = wmma4(a, b, accv[nt]);
    }
  }
  // gate and park G in per-wave LDS (row-major 16x64, pitch 65)
  float* gb = lds_g + w * (16 * 65);
#pragma unroll
  for (int nt = 0; nt < 4; ++nt) {
    int h = nt >> 1;                             // d = nt*16+l16; head = d/32
#pragma unroll
    for (int r = 0; r < 8; ++r) {
      int m = r + half * 8;
      gb[m * 65 + nt * 16 + l16] = accv[nt][r] * lds_kg[h * 64 + w * 16 + m];
    }
  }
  // out-tile: G(16x64) @ W_out^T
  v8f acco[4] = {{}, {}, {}, {}};
  for (int kc = 0; kc < 16; ++kc) {
    int kk = kc * 4 + half * 2;
    v2f a; a.x = gb[l16 * 65 + kk]; a.y = gb[l16 * 65 + kk + 1];
#pragma unroll
    for (int nt = 0; nt < 4; ++nt) {
      v2f b; b.x = lds_wo[kk * 65 + nt * 16 + l16];
      b.y = lds_wo[(kk + 1) * 65 + nt * 16 + l16];
      acco[nt] = wmma4(a, b, acco[nt]);
    }
  }
  // epilogue: + b_out, fold back to (bn, c, H, W)
  float* ob = out + (size_t)bn * CCH * IMG * IMG;
#pragma unroll
  for (int nt = 0; nt < 4; ++nt) {
#pragma unroll
    for (int r = 0; r < 8; ++r) {
      int m = r + half * 8;
      int p = nt * 16 + l16;
      int c = w * 16 + m;
      int row = py * PS + (p >> 3), col = px * PS + (p & 7);
      ob[((size_t)c * IMG + row) * IMG + col] = acco[nt][r] + bo[p];
    }
  }
}

// ---------------------------------------------------------------------------
extern "C" void kernel_launch(void* const* d_in, const int* in_sizes, int n_in,
                              void* d_out, int out_size, void* d_ws, size_t ws_size,
                              hipStream_t stream) {
  (void)in_sizes; (void)n_in; (void)out_size; (void)ws_size;
  const float* x    = (const float*)d_in[0];
  const float* Wemb = (const float*)d_in[1];
  const float* bemb = (const float*)d_in[2];
  const float* pos  = (const float*)d_in[3];
  const float* Wq   = (const float*)d_in[4];
  const float* Wk   = (const float*)d_in[5];
  const float* Wv   = (const float*)d_in[6];
  const float* Wo   = (const float*)d_in[7];
  const float* bo   = (const float*)d_in[8];
  float* out = (float*)d_out;

  float* ws  = (float*)d_ws;
  float* xe  = ws;                                 // 8*16384*64 floats (33.5 MB)
  float* Q   = xe + (size_t)BN * NPC * DIM;        // 16*16384
  float* KG  = Q + (size_t)QROWS * NPC;            // 16*16384
  float* WqA = KG + (size_t)QROWS * NPC;           // 128

  k_wqavg<<<1, 128, 0, stream>>>(Wq, WqA);
  k_embed<<<BN * NPATCH, 128, 0, stream>>>(x, Wemb, bemb, pos, xe);
  k_q<<<(BN * NPC) / 256, 256, 0, stream>>>(xe, WqA, Q);
  k_gate<<<NPC / 64, 128, 0, stream>>>(Q, Wk, KG);
  k_out<<<BN * NPATCH, 128, 0, stream>>>(xe, Wv, Wo, bo, KG, out);
}